// StackGraphAttention_4896262717822
// MI455X (gfx1250) — compile-verified
//
#include <hip/hip_runtime.h>

typedef __attribute__((ext_vector_type(2))) float v2f;
typedef __attribute__((ext_vector_type(8))) float v8f;

#define NN 4096
#define TT 32
#define DI 256
#define DO 128
#define HH 4
#define LDST 144   // LDS row stride (floats): 2*144 % 64 == 32 -> halves hit disjoint banks

__device__ __forceinline__ v8f wmma4(v2f a, v2f b, v8f c) {
  // D = A(16x4,f32) * B(4x16,f32) + C(16x16,f32)
  return __builtin_amdgcn_wmma_f32_16x16x4_f32(false, a, false, b, (short)0, c,
                                               false, false);
}

// ---------------------------------------------------------------------------
// Kernel 1: fingerprint projection  v = tanh(x@W + b), vu = v . u
// One wave per 16-row tile of the (N*T) x DI matrix; WMMA over K=256,
// fused tanh/bias/dot-u epilogue + cross-lane row-sum reduction.
// ---------------------------------------------------------------------------
__global__ void __launch_bounds__(256) fp_proj_vu_kernel(
    const float* __restrict__ x, const float* __restrict__ W,
    const float* __restrict__ bias, const float* __restrict__ u,
    float* __restrict__ vu) {
  const int lane = threadIdx.x & 31;
  const int wave = threadIdx.x >> 5;
  const int tile = blockIdx.x * 8 + wave;
  const int row0 = tile * 16;
  const int hk   = lane >> 4;   // which half of the wave
  const int l16  = lane & 15;
  const long arow = row0 + l16;

  float partial[8];
  for (int r = 0; r < 8; ++r) partial[r] = 0.f;

  for (int c = 0; c < 8; ++c) {
    v8f acc = {};
    const int col = c * 16 + l16;
    for (int k0 = 0; k0 < DI; k0 += 4) {
      const int kk = k0 + 2 * hk;
      v2f a; a.x = x[arow * DI + kk];      a.y = x[arow * DI + kk + 1];
      v2f b; b.x = W[(long)kk * DO + col]; b.y = W[(long)(kk + 1) * DO + col];
      acc = wmma4(a, b, acc);
    }
    const float bv = bias[col], uv = u[col];
    for (int r = 0; r < 8; ++r) partial[r] += tanhf(acc[r] + bv) * uv;
  }
  // reduce the 128 columns' contributions: sum across the 16 lanes of each half
  for (int r = 0; r < 8; ++r) {
    float s = partial[r];
    s += __shfl_xor(s, 1, 32); s += __shfl_xor(s, 2, 32);
    s += __shfl_xor(s, 4, 32); s += __shfl_xor(s, 8, 32);
    if (l16 == 0) vu[row0 + r + 8 * hk] = s;   // C row r+8*hk
  }
}

// ---------------------------------------------------------------------------
// Kernel 2: softmax over T per node + weighted sum of x  ->  xf [N, DI]
// One wave per node: lane t holds vu[n,t] (T == 32).
// ---------------------------------------------------------------------------
__global__ void __launch_bounds__(256) fp_softmax_xf_kernel(
    const float* __restrict__ x, const float* __restrict__ vu,
    float* __restrict__ xf) {
  const int lane = threadIdx.x & 31;
  const int wave = threadIdx.x >> 5;
  const int n = blockIdx.x * 8 + wave;
  float v = vu[n * TT + lane];
  float m = v;
  for (int d = 16; d > 0; d >>= 1) m = fmaxf(m, __shfl_xor(m, d, 32));
  float e = __expf(v - m);
  float s = e;
  for (int d = 16; d > 0; d >>= 1) s += __shfl_xor(s, d, 32);
  const float alpha = e / s;
  const float* xn = x + (long)n * TT * DI;
  for (int k0 = 0; k0 < DI; k0 += 32) {
    const int k = k0 + lane;
    float acc = 0.f;
    for (int t = 0; t < TT; ++t)
      acc += __shfl(alpha, t, 32) * xn[(long)t * DI + k];
    xf[(long)n * DI + k] = acc;
  }
}

// ---------------------------------------------------------------------------
// Kernel 3: per-head projection xs[h] = A[NN x K] @ B[h][K x DO]  (WMMA)
// One wave per 16-row tile; grid.y = head. B is <=256KB and L1/L2 resident.
// ---------------------------------------------------------------------------
__global__ void __launch_bounds__(256) gemm_proj_kernel(
    const float* __restrict__ A, const float* __restrict__ B,
    float* __restrict__ C, int K) {
  const int lane = threadIdx.x & 31;
  const int wave = threadIdx.x >> 5;
  const int tile = blockIdx.x * 8 + wave;
  const int h = blockIdx.y;
  const int row0 = tile * 16;
  const int hk  = lane >> 4;
  const int l16 = lane & 15;
  const long arow = row0 + l16;
  const float* Bh = B + (long)h * K * DO;
  float* Ch = C + (long)h * NN * DO;

  for (int c = 0; c < 8; ++c) {
    v8f acc = {};
    const int col = c * 16 + l16;
    for (int k0 = 0; k0 < K; k0 += 4) {
      const int kk = k0 + 2 * hk;
      v2f a; a.x = A[arow * (long)K + kk];  a.y = A[arow * (long)K + kk + 1];
      v2f b; b.x = Bh[(long)kk * DO + col]; b.y = Bh[(long)(kk + 1) * DO + col];
      acc = wmma4(a, b, acc);
    }
    for (int r = 0; r < 8; ++r)
      Ch[(long)(row0 + r + 8 * hk) * DO + col] = acc[r];
  }
}

// ---------------------------------------------------------------------------
// Kernel 4: f1[h,n] = xs[h,n,:].aw1[h] + ab1[h]   (and f2 likewise)
// ---------------------------------------------------------------------------
__global__ void __launch_bounds__(256) attn_feat_kernel(
    const float* __restrict__ xs, const float* __restrict__ aw1,
    const float* __restrict__ ab1, const float* __restrict__ aw2,
    const float* __restrict__ ab2, float* __restrict__ f1,
    float* __restrict__ f2) {
  const int i = blockIdx.x * blockDim.x + threadIdx.x;  // i = h*NN + n
  const int h = i / NN;
  const float* xr = xs + (long)i * DO;
  float s1 = ab1[h], s2 = ab2[h];
  for (int e = 0; e < DO; ++e) {
    const float xv = xr[e];
    s1 += xv * aw1[h * DO + e];
    s2 += xv * aw2[h * DO + e];
  }
  f1[i] = s1;
  f2[i] = s2;
}

// ---------------------------------------------------------------------------
// Kernel 5: streaming GAT attention row-block.
// scores[n,m] = leaky_relu(f1[n]+f2[m]) + adj[n,m]; softmax over m;
// out = softmax @ xs + ob; ELU.
// 8 waves/block sweep the same 16-column chunks -> the shared xs tile (8KB)
// and f2 chunk are double-buffered through LDS (one barrier per chunk,
// next-chunk global loads overlap the WMMA work). Online softmax, fp32 WMMA
// for the P @ xs accumulation. MODE 0: write head-concat. MODE 1: atomic mean.
// ---------------------------------------------------------------------------
template <int MODE>
__global__ void __launch_bounds__(256) gat_attn_kernel(
    const float* __restrict__ xs, const float* __restrict__ f1,
    const float* __restrict__ f2, const float* __restrict__ adj,
    const float* __restrict__ ob, float* __restrict__ out) {
  __shared__ float sxs[2][16 * LDST];
  __shared__ float sf2[2][16];

  const int tid  = threadIdx.x;
  const int lane = tid & 31;
  const int wave = tid >> 5;
  const int tile = blockIdx.x * 8 + wave;
  const int h = blockIdx.y;
  const int row0 = tile * 16;
  const int hk  = lane >> 4;
  const int l16 = lane & 15;
  const int prow = row0 + l16;                 // A-matrix row this lane owns

  const float f1r = f1[h * NN + prow];
  const float* __restrict__ f2h  = f2 + h * NN;
  const float* __restrict__ xsh  = xs + (long)h * NN * DO;
  const float* __restrict__ adjr = adj + (long)prow * NN;

  // cooperative-load mapping: thread t -> xs row t/16, cols (t%16)*8 .. +7
  const int ldr = tid >> 4;
  const int ldc = (tid & 15) * 8;

  float m = -1e30f, l = 0.f;
  v8f acc[8];
  {
    v8f z = {};
    for (int c = 0; c < 8; ++c) acc[c] = z;
  }

  // prefetch + stage chunk 0
  float4 pre0, pre1;
  float pref2 = 0.f;
  pre0 = *(const float4*)&xsh[(long)ldr * DO + ldc];
  pre1 = *(const float4*)&xsh[(long)ldr * DO + ldc + 4];
  if (tid < 16) pref2 = f2h[tid];
  *(float4*)&sxs[0][ldr * LDST + ldc]     = pre0;
  *(float4*)&sxs[0][ldr * LDST + ldc + 4] = pre1;
  if (tid < 16) sf2[0][tid] = pref2;

  int cur = 0;
  for (int j0 = 0; j0 < NN; j0 += 16) {
    const bool havenext = (j0 + 16 < NN);      // block-uniform
    if (havenext) {
      const long jn = j0 + 16;
      pre0 = *(const float4*)&xsh[(jn + ldr) * DO + ldc];
      pre1 = *(const float4*)&xsh[(jn + ldr) * DO + ldc + 4];
      if (tid < 16) pref2 = f2h[jn + tid];
    }
    __syncthreads();                           // sxs[cur]/sf2[cur] now visible

    // build the 16x16 score tile directly in A-fragment register layout:
    // lane holds row prow, columns 4t + 2*hk + {0,1}
    float p[8];
    float cmax = -1e30f;
    for (int t = 0; t < 4; ++t) {
      for (int s = 0; s < 2; ++s) {
        const int cc = 4 * t + 2 * hk + s;
        float v = f1r + sf2[cur][cc];
        v = (v > 0.f) ? v : 0.2f * v;           // leaky_relu(0.2)
        v += adjr[j0 + cc];
        p[2 * t + s] = v;
        cmax = fmaxf(cmax, v);
      }
    }
    cmax = fmaxf(cmax, __shfl_xor(cmax, 16, 32));  // lanes L, L^16 share a row
    const float mnew = fmaxf(m, cmax);
    const float scale = __expf(m - mnew);
    m = mnew;
    float csum = 0.f;
    for (int i = 0; i < 8; ++i) { p[i] = __expf(p[i] - mnew); csum += p[i]; }
    csum += __shfl_xor(csum, 16, 32);
    l = l * scale + csum;

    // rescale accumulators: C element r on this lane is row r+8*hk, whose
    // running max lives on lane (r+8*hk)
    float scr[8];
    for (int r = 0; r < 8; ++r) scr[r] = __shfl(scale, r + 8 * hk, 32);
    for (int c = 0; c < 8; ++c)
      for (int r = 0; r < 8; ++r) acc[c][r] *= scr[r];

    // P(16x16) @ xs_tile(16x128): 4 K-steps x 8 column tiles of fp32 WMMA,
    // B fragments served from LDS
    const float* xt = &sxs[cur][0];
    for (int t = 0; t < 4; ++t) {
      v2f a; a.x = p[2 * t]; a.y = p[2 * t + 1];
      const int kl = 4 * t + 2 * hk;
      for (int c = 0; c < 8; ++c) {
        const int col = c * 16 + l16;
        v2f b; b.x = xt[kl * LDST + col]; b.y = xt[(kl + 1) * LDST + col];
        acc[c] = wmma4(a, b, acc[c]);
      }
    }

    if (havenext) {
      const int nb = cur ^ 1;
      *(float4*)&sxs[nb][ldr * LDST + ldc]     = pre0;
      *(float4*)&sxs[nb][ldr * LDST + ldc + 4] = pre1;
      if (tid < 16) sf2[nb][tid] = pref2;
    }
    cur ^= 1;
  }

  float lr[8];
  for (int r = 0; r < 8; ++r) lr[r] = __shfl(l, r + 8 * hk, 32);
  for (int c = 0; c < 8; ++c) {
    const int col = c * 16 + l16;
    const float obv = ob[h * DO + col];
    for (int r = 0; r < 8; ++r) {
      float v = acc[c][r] / lr[r] + obv;
      v = (v > 0.f) ? v : (__expf(v) - 1.f);    // ELU
      const long row = row0 + r + 8 * hk;
      if (MODE == 0)
        out[row * (HH * DO) + h * DO + col] = v;         // head-concat layout
      else
        atomicAdd(&out[row * DO + col], 0.25f * v);      // mean over 4 heads
    }
  }
}

// ---------------------------------------------------------------------------
// Kernel 6: gather node embeddings
// ---------------------------------------------------------------------------
__global__ void __launch_bounds__(256) gather_kernel(
    const float* __restrict__ hm, const int* __restrict__ ids,
    float* __restrict__ out) {
  const int i = blockIdx.x * blockDim.x + threadIdx.x;   // 1024*128
  const int k = i >> 7;
  const int e = i & (DO - 1);
  out[i] = hm[(long)ids[k] * DO + e];
}

extern "C" void kernel_launch(void* const* d_in, const int* in_sizes, int n_in,
                              void* d_out, int out_size, void* d_ws,
                              size_t ws_size, hipStream_t stream) {
  (void)in_sizes; (void)n_in; (void)out_size; (void)ws_size;
  const float* x    = (const float*)d_in[0];
  const float* adj  = (const float*)d_in[1];
  const int*   ids  = (const int*)d_in[2];
  const float* Ww   = (const float*)d_in[3];
  const float* Wb   = (const float*)d_in[4];
  const float* Wu   = (const float*)d_in[5];
  const float* Wp0  = (const float*)d_in[6];
  const float* f1w0 = (const float*)d_in[7];
  const float* f1b0 = (const float*)d_in[8];
  const float* f2w0 = (const float*)d_in[9];
  const float* f2b0 = (const float*)d_in[10];
  const float* ob0  = (const float*)d_in[11];
  const float* Wp1  = (const float*)d_in[12];
  const float* f1w1 = (const float*)d_in[13];
  const float* f1b1 = (const float*)d_in[14];
  const float* f2w1 = (const float*)d_in[15];
  const float* f2b1 = (const float*)d_in[16];
  const float* ob1  = (const float*)d_in[17];
  float* out = (float*)d_out;

  float* ws    = (float*)d_ws;
  float* vu    = ws;  ws += (long)NN * TT;        // 0.5 MB
  float* xf    = ws;  ws += (long)NN * DI;        // 4 MB
  float* xs0   = ws;  ws += (long)HH * NN * DO;   // 8 MB
  float* f1a   = ws;  ws += (long)HH * NN;
  float* f2a   = ws;  ws += (long)HH * NN;
  float* hcat  = ws;  ws += (long)NN * HH * DO;   // 8 MB
  float* xs1   = ws;  ws += (long)HH * NN * DO;   // 8 MB
  float* f1c   = ws;  ws += (long)HH * NN;
  float* f2c   = ws;  ws += (long)HH * NN;
  float* hmean = ws;  ws += (long)NN * DO;        // 2 MB

  const dim3 blk(256);

  // Stage 1: fingerprint attention  ->  xf [N, 256]
  fp_proj_vu_kernel<<<dim3((NN * TT) / 128), blk, 0, stream>>>(x, Ww, Wb, Wu, vu);
  fp_softmax_xf_kernel<<<dim3(NN / 8), blk, 0, stream>>>(x, vu, xf);

  // Stage 2: GAT layer 0  ->  hcat [N, H*128]
  gemm_proj_kernel<<<dim3(NN / 128, HH), blk, 0, stream>>>(xf, Wp0, xs0, DI);
  attn_feat_kernel<<<dim3((HH * NN) / 256), blk, 0, stream>>>(xs0, f1w0, f1b0,
                                                              f2w0, f2b0, f1a, f2a);
  gat_attn_kernel<0><<<dim3(NN / 128, HH), blk, 0, stream>>>(xs0, f1a, f2a, adj,
                                                             ob0, hcat);

  // Stage 3: GAT layer 1  ->  hmean [N, 128]
  gemm_proj_kernel<<<dim3(NN / 128, HH), blk, 0, stream>>>(hcat, Wp1, xs1, HH * DO);
  attn_feat_kernel<<<dim3((HH * NN) / 256), blk, 0, stream>>>(xs1, f1w1, f1b1,
                                                              f2w1, f2b1, f1c, f2c);
  hipMemsetAsync(hmean, 0, (size_t)NN * DO * sizeof(float), stream);
  gat_attn_kernel<1><<<dim3(NN / 128, HH), blk, 0, stream>>>(xs1, f1c, f2c, adj,
                                                             ob1, hmean);

  // Stage 4: gather 1024 node embeddings
  gather_kernel<<<dim3((1024 * DO) / 256), blk, 0, stream>>>(hmean, ids, out);
}